// NetVLAD_13812614824638
// MI455X (gfx1250) — compile-verified
//
#include <hip/hip_runtime.h>
#include <hip/hip_bf16.h>
#include <math.h>

typedef __attribute__((ext_vector_type(2))) float v2f;
typedef __attribute__((ext_vector_type(8))) float v8f;

#define EPSV 1e-12f
#define BATCH 64
#define HW    784
#define DDIM  512
#define KCL   64
#define MROWS (BATCH*HW)   /* 50176 */

// ---------------------------------------------------------------------------
// CDNA5 async global->LDS copy (ASYNCcnt-tracked), saddr addressing mode:
//   LDS address from the vdst VGPR; global address = SGPR-pair base + VGPR byte offset
// ---------------------------------------------------------------------------
__device__ __forceinline__ void async_copy_b128(float* lds_ptr,
                                                const float* gbase,
                                                unsigned voff_bytes) {
  unsigned loff = (unsigned)(uintptr_t)lds_ptr;  // low 32 bits = LDS offset
  unsigned long long base = (unsigned long long)(uintptr_t)gbase;
  asm volatile("global_load_async_to_lds_b128 %0, %1, %2"
               :: "v"(loff), "v"(voff_bytes), "s"(base)
               : "memory");
}

// ---------------------------------------------------------------------------
// zero the atomic accumulators (a_sum + norms)
// ---------------------------------------------------------------------------
__global__ void zero_ws(float* p, int n) {
  int i = blockIdx.x * blockDim.x + threadIdx.x;
  if (i < n) p[i] = 0.f;
}

// ---------------------------------------------------------------------------
// Phase 1: s = x*kernel + bias ; softmax over K ; a -> ws ; a_sum (atomic)
// Block: 256 threads = 8 waves; wave owns 16 rows x 64 cols (4 WMMA N-tiles).
// x tile staged [row][d] (stride 68); kernel slice staged TRANSPOSED [k][d]
// (stride 68) so both A and B fragments are single conflict-free ds_load_b64.
// ---------------------------------------------------------------------------
__global__ __launch_bounds__(256) void phase1_assign(
    const float* __restrict__ x, const float* __restrict__ kern,
    const float* __restrict__ bias, float* __restrict__ a_ws,
    float* __restrict__ asum) {
  __shared__ float xs[128 * 68];   // 34.0 KB
  __shared__ float kst[64 * 68];   // 17.0 KB, transposed kernel slice

  const int tid    = threadIdx.x;
  const int lane   = tid & 31;
  const int wave   = tid >> 5;
  const int lanelo = lane & 15;
  const int kpair  = (lane >> 4) << 1;   // lanes 0-15 -> K 0/1, 16-31 -> K 2/3
  const int row0   = blockIdx.x * 128;

  v8f acc[4];
#pragma unroll
  for (int t = 0; t < 4; ++t) acc[t] = (v8f){0.f,0.f,0.f,0.f,0.f,0.f,0.f,0.f};

  for (int dd0 = 0; dd0 < DDIM; dd0 += 64) {
    __syncthreads();
    // stage x tile [128 x 64] (2048 float4, 8/thread), coalesced
#pragma unroll
    for (int i = 0; i < 8; ++i) {
      int idx = tid + i * 256;
      int r = idx >> 4, c4 = idx & 15;
      float4 v = *(const float4*)(x + (size_t)(row0 + r) * DDIM + dd0 + c4 * 4);
      *(float4*)(xs + r * 68 + c4 * 4) = v;
    }
    // stage kernel slice transposed: read [d][k] coalesced, write [k][d]
#pragma unroll
    for (int i = 0; i < 4; ++i) {
      int idx = tid + i * 256;          // 0..1023
      int r = idx >> 4, c4 = idx & 15;  // r = local d, c4 = k/4
      float4 v = *(const float4*)(kern + (size_t)(dd0 + r) * KCL + c4 * 4);
      kst[(c4 * 4 + 0) * 68 + r] = v.x;
      kst[(c4 * 4 + 1) * 68 + r] = v.y;
      kst[(c4 * 4 + 2) * 68 + r] = v.z;
      kst[(c4 * 4 + 3) * 68 + r] = v.w;
    }
    __syncthreads();

    const int arow = wave * 16 + lanelo;
#pragma unroll 4
    for (int dl = 0; dl < 64; dl += 4) {
      v2f af = *(const v2f*)(xs + arow * 68 + dl + kpair);        // ds_load_b64
#pragma unroll
      for (int t = 0; t < 4; ++t) {
        v2f bf = *(const v2f*)(kst + (t * 16 + lanelo) * 68 + dl + kpair);
        acc[t] = __builtin_amdgcn_wmma_f32_16x16x4_f32(
            false, af, false, bf, (short)0, acc[t], false, false);
      }
    }
  }

  // ---- softmax over K=64 within the wave ----
  // C/D layout: VGPR j: lanes 0-15 -> row j, lanes 16-31 -> row j+8; col = lanelo
  const int rowbase = row0 + wave * 16;
  const int b = rowbase / HW;              // 16-row tiles never cross a batch (784 = 49*16)
  const int rhalf = (lane >> 4) * 8;
  float bv[4];
#pragma unroll
  for (int t = 0; t < 4; ++t) bv[t] = bias[t * 16 + lanelo];

  float part0 = 0.f, part1 = 0.f, part2 = 0.f, part3 = 0.f;
#pragma unroll
  for (int j = 0; j < 8; ++j) {
    float s0 = acc[0][j] + bv[0];
    float s1 = acc[1][j] + bv[1];
    float s2 = acc[2][j] + bv[2];
    float s3 = acc[3][j] + bv[3];
    float m = fmaxf(fmaxf(s0, s1), fmaxf(s2, s3));
    m = fmaxf(m, __shfl_xor(m, 1));
    m = fmaxf(m, __shfl_xor(m, 2));
    m = fmaxf(m, __shfl_xor(m, 4));
    m = fmaxf(m, __shfl_xor(m, 8));     // reduces within each 16-lane half
    float e0 = __expf(s0 - m), e1 = __expf(s1 - m);
    float e2 = __expf(s2 - m), e3 = __expf(s3 - m);
    float sum = e0 + e1 + e2 + e3;
    sum += __shfl_xor(sum, 1);
    sum += __shfl_xor(sum, 2);
    sum += __shfl_xor(sum, 4);
    sum += __shfl_xor(sum, 8);
    float r = 1.0f / sum;
    e0 *= r; e1 *= r; e2 *= r; e3 *= r;
    float* ap = a_ws + (size_t)(rowbase + j + rhalf) * KCL + lanelo;
    ap[0] = e0; ap[16] = e1; ap[32] = e2; ap[48] = e3;
    part0 += e0; part1 += e1; part2 += e2; part3 += e3;
  }
  atomicAdd(asum + b * KCL +  0 + lanelo, part0);
  atomicAdd(asum + b * KCL + 16 + lanelo, part1);
  atomicAdd(asum + b * KCL + 32 + lanelo, part2);
  atomicAdd(asum + b * KCL + 48 + lanelo, part3);
}

// ---------------------------------------------------------------------------
// Phase 2: v[b,k,d] = sum_hw a[hw,k]*x[hw,d] + C[d,k]*a_sum[b,k]
// Grid (8 d-tiles, 64 batches); 8 waves: 4 k-tiles x 2 d-halves of 32.
// x/a chunks stream through double-buffered LDS via GLOBAL_LOAD_ASYNC_TO_LDS
// (each thread moves one b128 slice; ASYNCcnt-tracked, overlap with WMMA).
// ---------------------------------------------------------------------------
__global__ __launch_bounds__(256) void phase2_vlad(
    const float* __restrict__ x, const float* __restrict__ a_ws,
    const float* __restrict__ asum, const float* __restrict__ Cmat,
    float* __restrict__ v_ws, float* __restrict__ norms) {
  __shared__ float xs[2][16 * 64];
  __shared__ float as[2][16 * 64];

  const int tid    = threadIdx.x;
  const int lane   = tid & 31;
  const int wave   = tid >> 5;
  const int lanelo = lane & 15;
  const int kpair  = (lane >> 4) << 1;
  const int b      = blockIdx.y;
  const int dbase  = blockIdx.x * 64;
  const int kt     = (wave & 3) * 16;
  const int dsub   = (wave >> 2) * 32;
  const int NC     = HW / 16;            // 49 chunks of 16 hw positions

  // per-thread b128 slice of each 16x64 chunk
  const int r  = tid >> 4;               // hw row 0..15
  const int c4 = tid & 15;               // float4 column
  const unsigned voff_x = (unsigned)((r * DDIM + c4 * 4) * sizeof(float));
  const unsigned voff_a = (unsigned)((r * KCL  + c4 * 4) * sizeof(float));
  const float* gx = x    + (size_t)b * HW * DDIM + dbase;
  const float* ga = a_ws + (size_t)b * HW * KCL;
  float* const ldsx[2] = { &xs[0][r * 64 + c4 * 4], &xs[1][r * 64 + c4 * 4] };
  float* const ldsa[2] = { &as[0][r * 64 + c4 * 4], &as[1][r * 64 + c4 * 4] };

  v8f acc[2];
  acc[0] = (v8f){0.f,0.f,0.f,0.f,0.f,0.f,0.f,0.f};
  acc[1] = (v8f){0.f,0.f,0.f,0.f,0.f,0.f,0.f,0.f};

  // prologue: chunk 0 in flight
  async_copy_b128(ldsx[0], gx, voff_x);
  async_copy_b128(ldsa[0], ga, voff_a);

  for (int c = 0; c < NC; ++c) {
    if (c + 1 < NC) {   // issue next chunk, then wait for current (<=2 newer ops)
      async_copy_b128(ldsx[(c + 1) & 1], gx + (size_t)(c + 1) * 16 * DDIM, voff_x);
      async_copy_b128(ldsa[(c + 1) & 1], ga + (size_t)(c + 1) * 16 * KCL,  voff_a);
      asm volatile("s_wait_asynccnt 0x2" ::: "memory");
    } else {
      asm volatile("s_wait_asynccnt 0x0" ::: "memory");
    }
    __syncthreads();

    const float* xb = xs[c & 1];
    const float* ab = as[c & 1];
#pragma unroll
    for (int hi = 0; hi < 16; hi += 4) {
      v2f af;   // A[m=k][kk=hw] = a[hw][k]
      af.x = ab[(hi + kpair + 0) * 64 + kt + lanelo];
      af.y = ab[(hi + kpair + 1) * 64 + kt + lanelo];
#pragma unroll
      for (int t = 0; t < 2; ++t) {
        v2f bf; // B[kk=hw][n=d] = x[hw][d]
        bf.x = xb[(hi + kpair + 0) * 64 + dsub + t * 16 + lanelo];
        bf.y = xb[(hi + kpair + 1) * 64 + dsub + t * 16 + lanelo];
        acc[t] = __builtin_amdgcn_wmma_f32_16x16x4_f32(
            false, af, false, bf, (short)0, acc[t], false, false);
      }
    }
    __syncthreads();  // chunk consumed; its buffer may be re-filled next iter
  }

  // epilogue: C*a_sum term, store v (coalesced over d), accumulate norms
  const int rhalf = (lane >> 4) * 8;
#pragma unroll
  for (int j = 0; j < 8; ++j) {
    const int k = kt + j + rhalf;
    const float av = asum[b * KCL + k];
    float p = 0.f;
#pragma unroll
    for (int t = 0; t < 2; ++t) {
      const int d = dbase + dsub + t * 16 + lanelo;
      float v = acc[t][j] + Cmat[(size_t)d * KCL + k] * av;
      v_ws[((size_t)b * KCL + k) * DDIM + d] = v;
      p += v * v;
    }
    p += __shfl_xor(p, 1);
    p += __shfl_xor(p, 2);
    p += __shfl_xor(p, 4);
    p += __shfl_xor(p, 8);
    if (lanelo == 0) atomicAdd(norms + b * KCL + k, p);
  }
}

// ---------------------------------------------------------------------------
// Phase 3: intra-norm (over D per cluster) * global norm, transpose to
// out[b, d*K + k] with LDS (stride-65 -> bank-conflict-free).
// ---------------------------------------------------------------------------
__global__ __launch_bounds__(256) void phase3_norm(
    const float* __restrict__ v_ws, const float* __restrict__ norms,
    float* __restrict__ out) {
  __shared__ float scale[KCL];
  __shared__ float term[KCL];
  __shared__ float tr[64 * 65];
  const int tid   = threadIdx.x;
  const int b     = blockIdx.y;
  const int dbase = blockIdx.x * 64;

  if (tid < KCL) {
    float n = norms[b * KCL + tid];
    term[tid]  = n / (n + EPSV);      // |v_k/||v_k|||^2 contribution
    scale[tid] = rsqrtf(n + EPSV);
  }
  __syncthreads();
  if (tid < KCL) {
    float g = 0.f;
    for (int k = 0; k < KCL; ++k) g += term[k];
    scale[tid] *= rsqrtf(g + EPSV);   // fold global L2 into per-cluster scale
  }
  __syncthreads();

#pragma unroll
  for (int i = 0; i < 16; ++i) {
    int idx = tid + i * 256;          // 0..4095
    int k = idx >> 6, d = idx & 63;   // coalesced read over d
    tr[d * 65 + k] = v_ws[((size_t)b * KCL + k) * DDIM + dbase + d] * scale[k];
  }
  __syncthreads();
#pragma unroll
  for (int i = 0; i < 16; ++i) {
    int idx = tid + i * 256;
    int d = idx >> 6, k = idx & 63;   // coalesced write over k
    out[(size_t)b * (DDIM * KCL) + (size_t)(dbase + d) * KCL + k] = tr[d * 65 + k];
  }
}

// ---------------------------------------------------------------------------
extern "C" void kernel_launch(void* const* d_in, const int* in_sizes, int n_in,
                              void* d_out, int out_size, void* d_ws, size_t ws_size,
                              hipStream_t stream) {
  const float* x    = (const float*)d_in[0];
  const float* kern = (const float*)d_in[1];
  const float* bias = (const float*)d_in[2];
  const float* Cmat = (const float*)d_in[3];
  float* out = (float*)d_out;

  float* ws    = (float*)d_ws;
  float* a_ws  = ws;                              // 50176*64 = 12.85 MB
  float* asum  = a_ws + (size_t)MROWS * KCL;      // 4096
  float* norms = asum + BATCH * KCL;              // 4096
  float* v_ws  = norms + BATCH * KCL;             // 64*64*512 = 8.39 MB

  zero_ws<<<32, 256, 0, stream>>>(asum, 2 * BATCH * KCL);
  phase1_assign<<<MROWS / 128, 256, 0, stream>>>(x, kern, bias, a_ws, asum);
  phase2_vlad<<<dim3(DDIM / 64, BATCH), 256, 0, stream>>>(x, a_ws, asum, Cmat,
                                                          v_ws, norms);
  phase3_norm<<<dim3(DDIM / 64, BATCH), 256, 0, stream>>>(v_ws, norms, out);
}